// HistogramLayer_28037546508489
// MI455X (gfx1250) — compile-verified
//
#include <hip/hip_runtime.h>

// HistogramLayer: x (B=16, C=10, H=512, W=512) fp32
//   cos = x[:, :8], grad = x[:, 8:10]
//   norm = sqrt(gx^2 + gy^2); idx = argmax_c cos
//   out[b, c, p] = (c == idx) ? norm : 0        -> (16, 8, 512, 512) fp32
//
// Pure streaming kernel: 72 B/pixel, ~30 flops/pixel -> HBM-bound (~13 us floor
// at 23.3 TB/s). Optimized for gfx1250 via wave32-coalesced B128 vector memory
// ops with non-temporal (TH=NT) cache hints (single-touch data, in+out > 192MB L2).

typedef float v4f __attribute__((ext_vector_type(4)));

#define NUM_BINS 8
#define TOTAL_CH 10
#define PLANE (512 * 512)          // H*W
#define VEC 4
#define NVEC_PER_BATCH (PLANE / VEC)   // 65536 = 1 << 16

__global__ __launch_bounds__(256) void hist_bins_kernel(
    const float* __restrict__ x, float* __restrict__ out)
{
    // One thread handles 4 consecutive pixels of one batch image.
    const size_t t   = (size_t)blockIdx.x * blockDim.x + threadIdx.x;
    const size_t b   = t >> 16;                 // / NVEC_PER_BATCH
    const size_t pv  = t & (NVEC_PER_BATCH - 1);
    const size_t pix = pv * VEC;

    const float* xb = x + b * (size_t)TOTAL_CH * PLANE + pix;

    // 10 coalesced B128 non-temporal loads (one per channel plane).
    v4f cosv[NUM_BINS];
#pragma unroll
    for (int c = 0; c < NUM_BINS; ++c)
        cosv[c] = __builtin_nontemporal_load((const v4f*)(xb + (size_t)c * PLANE));
    const v4f gx = __builtin_nontemporal_load((const v4f*)(xb + (size_t)NUM_BINS * PLANE));
    const v4f gy = __builtin_nontemporal_load((const v4f*)(xb + (size_t)(NUM_BINS + 1) * PLANE));

    // Per-pixel argmax (strict '>' keeps first max, matching jnp.argmax) + norm.
    int bi[VEC];
    v4f nrm;
#pragma unroll
    for (int j = 0; j < VEC; ++j) {
        float best = cosv[0][j];
        int   idx  = 0;
#pragma unroll
        for (int c = 1; c < NUM_BINS; ++c) {
            if (cosv[c][j] > best) { best = cosv[c][j]; idx = c; }
        }
        bi[j]  = idx;
        nrm[j] = sqrtf(gx[j] * gx[j] + gy[j] * gy[j]);
    }

    // 8 coalesced B128 non-temporal stores.
    float* ob = out + b * (size_t)NUM_BINS * PLANE + pix;
#pragma unroll
    for (int c = 0; c < NUM_BINS; ++c) {
        v4f o;
#pragma unroll
        for (int j = 0; j < VEC; ++j)
            o[j] = (bi[j] == c) ? nrm[j] : 0.0f;
        __builtin_nontemporal_store(o, (v4f*)(ob + (size_t)c * PLANE));
    }
}

extern "C" void kernel_launch(void* const* d_in, const int* in_sizes, int n_in,
                              void* d_out, int out_size, void* d_ws, size_t ws_size,
                              hipStream_t stream) {
    const float* x   = (const float*)d_in[0];
    float*       out = (float*)d_out;

    const int batch    = in_sizes[0] / (TOTAL_CH * PLANE);   // 16
    const long long nt = (long long)batch * NVEC_PER_BATCH;  // 1,048,576 threads
    const int block    = 256;                                // 8 wave32 waves
    const int grid     = (int)((nt + block - 1) / block);    // 4096 blocks

    hist_bins_kernel<<<grid, block, 0, stream>>>(x, out);
}